// RelationalTrajectoryModel_61314953118351
// MI455X (gfx1250) — compile-verified
//
#include <hip/hip_runtime.h>
#include <hip/hip_bf16.h>

typedef __attribute__((ext_vector_type(2))) float v2f;
typedef __attribute__((ext_vector_type(8))) float v8f;

__device__ __forceinline__ v8f wmma4(v2f a, v2f b, v8f c) {
  // V_WMMA_F32_16X16X4_F32 : D = A(16x4) * B(4x16) + C(16x16), fp32 exact
  return __builtin_amdgcn_wmma_f32_16x16x4_f32(false, a, false, b, (short)0, c, false, false);
}

__device__ __forceinline__ float gelu_exact(float u) {
  return 0.5f * u * (1.0f + erff(u * 0.70710678118654752f));
}

// ---------------------------------------------------------------- node proj
// x[bt*128+n][c] = sum_{j<16} nf[row][j] * W[j][c] + b[c]
__global__ void k_nodeproj(const float* __restrict__ nf, const float* __restrict__ W,
                           const float* __restrict__ b, float* __restrict__ X) {
  int idx = blockIdx.x * 256 + threadIdx.x;     // 524288 total
  int row = idx >> 7, col = idx & 127;
  const float* r = nf + row * 16;
  float acc = b[col];
#pragma unroll
  for (int j = 0; j < 16; ++j) acc += r[j] * W[j * 128 + col];
  X[idx] = acc;
}

// ---------------------------------------------------------------- W2 head-mean
__global__ void k_prep(const float* __restrict__ W2, const float* __restrict__ b2,
                       float* __restrict__ W2m, float* __restrict__ b2m) {
  int c = threadIdx.x;                          // 128
#pragma unroll
  for (int h = 0; h < 4; ++h) {
    float s = 0.f;
    for (int d = 0; d < 32; ++d) s += W2[c * 128 + h * 32 + d];
    W2m[c * 4 + h] = s * (1.f / 32.f);
  }
  if (c < 4) {
    float s = 0.f;
    for (int d = 0; d < 32; ++d) s += b2[c * 32 + d];
    b2m[c] = s * (1.f / 32.f);
  }
}

// ---------------------------------------------------------------- batched GEMM (QKV)
// out[bt] = X[bt](128x128) @ W(128x128), one wave per 16x16 tile
__global__ void k_gemm(const float* __restrict__ X, const float* __restrict__ W,
                       float* __restrict__ out) {
  int nj = blockIdx.x, mi = blockIdx.y, bt = blockIdx.z;
  int lane = threadIdx.x & 31, half = lane >> 4, r = lane & 15;
  int row0 = mi * 16, col0 = nj * 16;
  const float* A = X + bt * 16384;
  v8f c = {};
  for (int kk = 0; kk < 128; kk += 4) {
    v2f a, b;
    const float* ar = A + (row0 + r) * 128 + kk + half * 2;
    a.x = ar[0]; a.y = ar[1];
    const float* br = W + (kk + half * 2) * 128 + col0 + r;
    b.x = br[0]; b.y = br[128];
    c = wmma4(a, b, c);
  }
  float* O = out + bt * 16384;
#pragma unroll
  for (int v = 0; v < 8; ++v)
    O[(row0 + v + half * 8) * 128 + col0 + r] = c[v];
}

// ---------------------------------------------------------------- geometric bias
// bias[bt][h][i][j] = gelu(geom(i,j)@W1 + b1) . W2m[:,h] + b2m[h]
__global__ void k_bias(const float* __restrict__ pos, const float* __restrict__ W1,
                       const float* __restrict__ b1, const float* __restrict__ W2m,
                       const float* __restrict__ b2m, float* __restrict__ bias) {
  int i = blockIdx.x, bt = blockIdx.y, j = threadIdx.x;   // 128 threads
  const float* pb = pos + bt * 256;
  float dx = pb[i * 2] - pb[j * 2];
  float dy = pb[i * 2 + 1] - pb[j * 2 + 1];
  float ss = dx * dx + dy * dy;
  float dist = sqrtf(ss + 1e-6f);
  float nrm = sqrtf(fmaxf(ss, 1e-12f));
  float g0 = dx / nrm, g1 = dy / nrm;
  float a0 = 0.f, a1 = 0.f, a2 = 0.f, a3 = 0.f;
  for (int c = 0; c < 128; ++c) {
    float u = g0 * W1[c] + g1 * W1[128 + c] + dist * W1[256 + c] + b1[c];
    float ge = gelu_exact(u);
    a0 += ge * W2m[c * 4 + 0];
    a1 += ge * W2m[c * 4 + 1];
    a2 += ge * W2m[c * 4 + 2];
    a3 += ge * W2m[c * 4 + 3];
  }
  float* bb = bias + ((bt * 4) * 128 + i) * 128 + j;
  bb[0 * 16384] = a0 + b2m[0];
  bb[1 * 16384] = a1 + b2m[1];
  bb[2 * 16384] = a2 + b2m[2];
  bb[3 * 16384] = a3 + b2m[3];
}

// ---------------------------------------------------------------- attention per (bt, head)
__global__ void k_attn(const float* __restrict__ Q, const float* __restrict__ K,
                       const float* __restrict__ V, const float* __restrict__ bias,
                       float* __restrict__ agg) {
  __shared__ float smem[128 * 128];                        // 64 KB score buffer
  int bt = blockIdx.x, h = blockIdx.y;
  int tid = threadIdx.x;                                   // 128 = 4 waves
  int wave = tid >> 5, lane = tid & 31, half = lane >> 4, r = lane & 15;
  const float* qb = Q + bt * 16384;
  const float* kb = K + bt * 16384;
  const float* vb = V + bt * 16384;
  const float* biasb = bias + ((bt * 4 + h) * 128) * 128;
  const float scl = 0.17677669529663687f;                  // 1/sqrt(32)

  // s = q k^T * scl + bias    (8x8 tiles of 16x16, K=32)
  for (int t = wave; t < 64; t += 4) {
    int row0 = (t >> 3) * 16, col0 = (t & 7) * 16;
    v8f c = {};
    for (int kk = 0; kk < 32; kk += 4) {
      v2f a, b;
      const float* ar = qb + (row0 + r) * 128 + h * 32 + kk + half * 2;
      a.x = ar[0]; a.y = ar[1];
      const float* br = kb + (col0 + r) * 128 + h * 32 + kk + half * 2;  // B = k^T
      b.x = br[0]; b.y = br[1];
      c = wmma4(a, b, c);
    }
#pragma unroll
    for (int v = 0; v < 8; ++v) {
      int row = row0 + v + half * 8, col = col0 + r;
      smem[row * 128 + col] = c[v] * scl + biasb[row * 128 + col];
    }
  }
  __syncthreads();

  // row softmax (one row per thread)
  {
    float* row = smem + tid * 128;
    float m = row[0];
    for (int j = 1; j < 128; ++j) m = fmaxf(m, row[j]);
    float sum = 0.f;
    for (int j = 0; j < 128; ++j) { float e = expf(row[j] - m); row[j] = e; sum += e; }
    float rs = 1.f / sum;
    for (int j = 0; j < 128; ++j) row[j] *= rs;
  }
  __syncthreads();

  // agg = a @ v_head   (8x2 tiles, K=128)
  for (int t = wave; t < 16; t += 4) {
    int row0 = (t >> 1) * 16, col0 = (t & 1) * 16;
    v8f c = {};
    for (int kk = 0; kk < 128; kk += 4) {
      v2f a, b;
      const float* ar = smem + (row0 + r) * 128 + kk + half * 2;
      a.x = ar[0]; a.y = ar[1];
      const float* br = vb + (kk + half * 2) * 128 + h * 32 + col0 + r;
      b.x = br[0]; b.y = br[128];
      c = wmma4(a, b, c);
    }
#pragma unroll
    for (int v = 0; v < 8; ++v) {
      int row = row0 + v + half * 8, col = col0 + r;
      agg[(bt * 128 + row) * 128 + h * 32 + col] = c[v];
    }
  }
}

// ---------------------------------------------------------------- out-proj + residual + LN (in-place on X)
__global__ void k_out_ln(const float* __restrict__ agg, const float* __restrict__ Wo,
                         const float* __restrict__ bo, const float* __restrict__ gam,
                         const float* __restrict__ bet, float* __restrict__ X) {
  __shared__ float outt[16 * 128];
  int mi = blockIdx.x, bt = blockIdx.y;
  int tid = threadIdx.x;                                   // 256 = 8 waves
  int wave = tid >> 5, lane = tid & 31, half = lane >> 4, r = lane & 15;
  int row0 = mi * 16, col0 = wave * 16;
  const float* A = agg + bt * 16384;
  v8f c = {};
  for (int kk = 0; kk < 128; kk += 4) {
    v2f a, b;
    const float* ar = A + (row0 + r) * 128 + kk + half * 2;
    a.x = ar[0]; a.y = ar[1];
    const float* br = Wo + (kk + half * 2) * 128 + col0 + r;
    b.x = br[0]; b.y = br[128];
    c = wmma4(a, b, c);
  }
#pragma unroll
  for (int v = 0; v < 8; ++v) {
    int rl = v + half * 8, col = col0 + r;
    outt[rl * 128 + col] = c[v] + bo[col];
  }
  __syncthreads();
  if (tid < 16) {
    float* xr = X + (bt * 128 + row0 + tid) * 128;
    const float* orow = outt + tid * 128;
    float s1 = 0.f, s2 = 0.f;
    for (int cc = 0; cc < 128; ++cc) { float t = xr[cc] + orow[cc]; s1 += t; s2 += t * t; }
    float mean = s1 * (1.f / 128.f);
    float inv = rsqrtf(s2 * (1.f / 128.f) - mean * mean + 1e-5f);
    for (int cc = 0; cc < 128; ++cc) {
      float t = xr[cc] + orow[cc];
      xr[cc] = (t - mean) * inv * gam[cc] + bet[cc];
    }
  }
}

// ---------------------------------------------------------------- pool + pos_emb
__global__ void k_pool(const float* __restrict__ X, const float* __restrict__ pe,
                       float* __restrict__ H) {
  int bt = blockIdx.x, c = threadIdx.x;
  float s = 0.f;
  for (int n = 0; n < 128; ++n) s += X[(bt * 128 + n) * 128 + c];
  H[bt * 128 + c] = s * (1.f / 128.f) + pe[(bt & 7) * 128 + c];
}

// ---------------------------------------------------------------- rel_seq copy
__global__ void k_copy(const float* __restrict__ X, float* __restrict__ dst) {
  int idx = blockIdx.x * 256 + threadIdx.x;
  dst[idx] = X[idx];
}

// ---------------------------------------------------------------- temporal encoder layer (tiny: T=8, D=128)
__global__ void k_enc(float* __restrict__ H,
                      const float* __restrict__ Wi, const float* __restrict__ bi,
                      const float* __restrict__ Wo, const float* __restrict__ bo,
                      const float* __restrict__ W1, const float* __restrict__ b1f,
                      const float* __restrict__ W2, const float* __restrict__ b2f,
                      const float* __restrict__ g1, const float* __restrict__ be1,
                      const float* __restrict__ g2, const float* __restrict__ be2) {
  __shared__ float hb[8 * 128];
  __shared__ float qkv[8 * 384];
  __shared__ float ao[8 * 128];
  __shared__ float x1[8 * 128];
  __shared__ float ff[8 * 512];
  int b = blockIdx.x, tid = threadIdx.x;                   // 256 threads
  float* hg = H + b * 1024;
  for (int idx = tid; idx < 1024; idx += 256) hb[idx] = hg[idx];
  __syncthreads();
  for (int idx = tid; idx < 8 * 384; idx += 256) {
    int t = idx / 384, c = idx % 384;
    float acc = bi[c];
    const float* hr = hb + t * 128;
    for (int d = 0; d < 128; ++d) acc += hr[d] * Wi[d * 384 + c];
    qkv[idx] = acc;
  }
  __syncthreads();
  if (tid < 32) {                                          // (head, t) pairs
    int hh = tid >> 3, t = tid & 7;
    const float scl = 0.17677669529663687f;
    float sc[8]; float m = -1e30f;
#pragma unroll
    for (int t2 = 0; t2 < 8; ++t2) {
      float acc = 0.f;
      for (int d = 0; d < 32; ++d)
        acc += qkv[t * 384 + hh * 32 + d] * qkv[t2 * 384 + 128 + hh * 32 + d];
      sc[t2] = acc * scl; m = fmaxf(m, sc[t2]);
    }
    float sum = 0.f;
#pragma unroll
    for (int t2 = 0; t2 < 8; ++t2) { sc[t2] = expf(sc[t2] - m); sum += sc[t2]; }
    float rs = 1.f / sum;
    for (int d = 0; d < 32; ++d) {
      float acc = 0.f;
#pragma unroll
      for (int t2 = 0; t2 < 8; ++t2) acc += sc[t2] * qkv[t2 * 384 + 256 + hh * 32 + d];
      ao[t * 128 + hh * 32 + d] = acc * rs;
    }
  }
  __syncthreads();
  for (int idx = tid; idx < 1024; idx += 256) {            // x1 = hb + ao@Wo + bo
    int t = idx >> 7, c = idx & 127;
    float acc = bo[c];
    const float* ar = ao + t * 128;
    for (int d = 0; d < 128; ++d) acc += ar[d] * Wo[d * 128 + c];
    x1[idx] = hb[idx] + acc;
  }
  __syncthreads();
  if (tid < 8) {                                           // LN1
    float* rw = x1 + tid * 128;
    float s1 = 0.f, s2 = 0.f;
    for (int c = 0; c < 128; ++c) { s1 += rw[c]; s2 += rw[c] * rw[c]; }
    float mean = s1 * (1.f / 128.f);
    float inv = rsqrtf(s2 * (1.f / 128.f) - mean * mean + 1e-5f);
    for (int c = 0; c < 128; ++c) rw[c] = (rw[c] - mean) * inv * g1[c] + be1[c];
  }
  __syncthreads();
  for (int idx = tid; idx < 8 * 512; idx += 256) {         // FFN up + relu
    int t = idx / 512, u = idx % 512;
    float acc = b1f[u];
    const float* rw = x1 + t * 128;
    for (int d = 0; d < 128; ++d) acc += rw[d] * W1[d * 512 + u];
    ff[idx] = fmaxf(acc, 0.f);
  }
  __syncthreads();
  for (int idx = tid; idx < 1024; idx += 256) {            // FFN down + residual
    int t = idx >> 7, c = idx & 127;
    float acc = b2f[c];
    const float* rw = ff + t * 512;
    for (int u = 0; u < 512; ++u) acc += rw[u] * W2[u * 128 + c];
    hb[idx] = x1[idx] + acc;
  }
  __syncthreads();
  if (tid < 8) {                                           // LN2 + writeback
    float* rw = hb + tid * 128;
    float s1 = 0.f, s2 = 0.f;
    for (int c = 0; c < 128; ++c) { s1 += rw[c]; s2 += rw[c] * rw[c]; }
    float mean = s1 * (1.f / 128.f);
    float inv = rsqrtf(s2 * (1.f / 128.f) - mean * mean + 1e-5f);
    for (int c = 0; c < 128; ++c) hg[tid * 128 + c] = (rw[c] - mean) * inv * g2[c] + be2[c];
  }
}

// ---------------------------------------------------------------- head
__global__ void k_head(const float* __restrict__ H, const float* __restrict__ g,
                       const float* __restrict__ be, const float* __restrict__ W1,
                       const float* __restrict__ b1, const float* __restrict__ W2,
                       const float* __restrict__ b2, float* __restrict__ y) {
  __shared__ float yn[128];
  __shared__ float z[128];
  __shared__ float mv[2];
  int bt = blockIdx.x, tid = threadIdx.x;                  // 128
  const float* hr = H + bt * 128;
  if (tid == 0) {
    float s1 = 0.f, s2 = 0.f;
    for (int c = 0; c < 128; ++c) { float t = hr[c]; s1 += t; s2 += t * t; }
    float mean = s1 * (1.f / 128.f);
    mv[0] = mean; mv[1] = rsqrtf(s2 * (1.f / 128.f) - mean * mean + 1e-5f);
  }
  __syncthreads();
  yn[tid] = (hr[tid] - mv[0]) * mv[1] * g[tid] + be[tid];
  __syncthreads();
  float acc = b1[tid];
  for (int c = 0; c < 128; ++c) acc += yn[c] * W1[c * 128 + tid];
  z[tid] = gelu_exact(acc);
  __syncthreads();
  if (tid < 2) {
    float a = b2[tid];
    for (int c = 0; c < 128; ++c) a += z[c] * W2[c * 2 + tid];
    y[bt * 2 + tid] = a;
  }
}

// ---------------------------------------------------------------- launcher
extern "C" void kernel_launch(void* const* d_in, const int* in_sizes, int n_in,
                              void* d_out, int out_size, void* d_ws, size_t ws_size,
                              hipStream_t stream) {
  (void)in_sizes; (void)n_in; (void)out_size; (void)ws_size;
  auto F = [&](int i) { return (const float*)d_in[i]; };
  const float* nf  = F(0);
  const float* pos = F(1);
  float* ws = (float*)d_ws;

  const size_t NMAT = 32u * 128u * 128u;                   // 524288
  float* X    = ws;                                        // [32][128][128]
  float* Q    = ws + 1 * NMAT;
  float* Kb   = ws + 2 * NMAT;
  float* V    = ws + 3 * NMAT;
  float* BIAS = ws + 4 * NMAT;                             // [32][4][128][128]
  float* AGG  = ws + 8 * NMAT;
  float* W2M  = ws + 9 * NMAT;                             // [3][128][4]
  float* B2M  = W2M + 3 * 512;                             // [3][4]
  float* H    = B2M + 64;                                  // [32][128]

  // node projection
  k_nodeproj<<<2048, 256, 0, stream>>>(nf, F(2), F(3), X);

  // per-layer W2 head means
  for (int l = 0; l < 3; ++l) {
    int g0 = 4 + 11 * l;   // Wq,Wk,Wv,W1,b1,W2,b2,Wo,bo,g,b
    k_prep<<<1, 128, 0, stream>>>(F(g0 + 5), F(g0 + 6), W2M + l * 512, B2M + l * 4);
  }

  // GNN layers
  for (int l = 0; l < 3; ++l) {
    int g0 = 4 + 11 * l;
    k_gemm<<<dim3(8, 8, 32), 32, 0, stream>>>(X, F(g0 + 0), Q);
    k_gemm<<<dim3(8, 8, 32), 32, 0, stream>>>(X, F(g0 + 1), Kb);
    k_gemm<<<dim3(8, 8, 32), 32, 0, stream>>>(X, F(g0 + 2), V);
    k_bias<<<dim3(128, 32), 128, 0, stream>>>(pos, F(g0 + 3), F(g0 + 4),
                                              W2M + l * 512, B2M + l * 4, BIAS);
    k_attn<<<dim3(32, 4), 128, 0, stream>>>(Q, Kb, V, BIAS, AGG);
    k_out_ln<<<dim3(8, 32), 256, 0, stream>>>(AGG, F(g0 + 7), F(g0 + 8),
                                              F(g0 + 9), F(g0 + 10), X);
  }

  // rel_seq output (second tuple element, after y's 64 floats)
  k_copy<<<2048, 256, 0, stream>>>(X, (float*)d_out + 64);

  // pool + pos_emb
  k_pool<<<32, 128, 0, stream>>>(X, F(37), H);

  // temporal encoder layers
  for (int l = 0; l < 2; ++l) {
    int b0 = 38 + 12 * l;  // Wi,bi,Wo,bo,W1,b1,W2,b2,g1,be1,g2,be2
    k_enc<<<4, 256, 0, stream>>>(H, F(b0 + 0), F(b0 + 1), F(b0 + 2), F(b0 + 3),
                                 F(b0 + 4), F(b0 + 5), F(b0 + 6), F(b0 + 7),
                                 F(b0 + 8), F(b0 + 9), F(b0 + 10), F(b0 + 11));
  }

  // head -> y (first 64 floats of d_out)
  k_head<<<32, 128, 0, stream>>>(H, F(62), F(63), F(64), F(65), F(66), F(67),
                                 (float*)d_out);
}